// CachedAttention_83949430768208
// MI455X (gfx1250) — compile-verified
//
#include <hip/hip_runtime.h>
#include <hip/hip_bf16.h>
#include <math.h>

// Problem constants (from reference)
#define T_TOK   2048
#define DMODEL  2048
#define H_Q     16
#define H_KV    8
#define D_K     128
#define S_CACHE 2048
#define RMS_EPS 1e-6f
#define QSCALE  0.08838834764831845f   // 1/sqrt(128)

typedef float v2f __attribute__((ext_vector_type(2)));
typedef float v8f __attribute__((ext_vector_type(8)));

// D(16x16,f32) = A(16x4,f32) * B(4x16,f32) + C
__device__ __forceinline__ v8f wmma_f32(v2f a, v2f b, v8f c) {
  return __builtin_amdgcn_wmma_f32_16x16x4_f32(
      /*neg_a=*/false, a, /*neg_b=*/false, b,
      /*c_mod=*/(short)0, c, /*reuse_a=*/false, /*reuse_b=*/false);
}

__device__ __forceinline__ float rowmax16(float v) {
  v = fmaxf(v, __shfl_xor(v, 1));
  v = fmaxf(v, __shfl_xor(v, 2));
  v = fmaxf(v, __shfl_xor(v, 4));
  v = fmaxf(v, __shfl_xor(v, 8));
  return v;
}
__device__ __forceinline__ float rowsum16(float v) {
  v += __shfl_xor(v, 1);
  v += __shfl_xor(v, 2);
  v += __shfl_xor(v, 4);
  v += __shfl_xor(v, 8);
  return v;
}

// ---------------------------------------------------------------------------
// Kernel 1: Q = x @ wq^T  -> RMSNorm -> *q_norm_w -> RoPE -> *1/sqrt(Dk)
// One wave per (16 tokens) x (one head's 128 dims).
// ---------------------------------------------------------------------------
__global__ void __launch_bounds__(32)
qproj_kernel(const float* __restrict__ x, const float* __restrict__ wq,
             const float* __restrict__ qw, float* __restrict__ qout) {
  const int lane  = threadIdx.x;
  const int tb    = blockIdx.x;            // token block (16 tokens)
  const int h     = blockIdx.y;            // query head
  const int t0    = tb * 16;
  const int n0    = h * D_K;               // feature-column base
  const int r     = lane & 15;             // A-row / B-col / C-col index
  const int koff  = (lane >> 4) * 2;       // K sub-offset per lane half
  const int ihalf = (lane >> 4) * 8;       // C/D row offset per lane half

  v8f acc[8] = {};                         // 16 x 128 f32 tile

  const float* xrow = x + (size_t)(t0 + r) * DMODEL;
  for (int kc = 0; kc < DMODEL / 4; ++kc) {
    const int k = kc * 4 + koff;
    v2f a = *(const v2f*)(xrow + k);
#pragma unroll
    for (int dc = 0; dc < 8; ++dc) {
      v2f b = *(const v2f*)(wq + (size_t)(n0 + dc * 16 + r) * DMODEL + k);
      acc[dc] = wmma_f32(a, b, acc[dc]);
    }
  }

  // --- RMSNorm over the 128 cols of each row ---
  float rstd[8];
#pragma unroll
  for (int i = 0; i < 8; ++i) {
    float ss = 0.f;
#pragma unroll
    for (int dc = 0; dc < 8; ++dc) ss += acc[dc][i] * acc[dc][i];
    ss = rowsum16(ss);
    rstd[i] = rsqrtf(ss * (1.f / (float)D_K) + RMS_EPS);
  }
#pragma unroll
  for (int dc = 0; dc < 8; ++dc) {
    float wv = qw[dc * 16 + r];
#pragma unroll
    for (int i = 0; i < 8; ++i) acc[dc][i] = acc[dc][i] * rstd[i] * wv;
  }

  // --- RoPE: pair (d, d+64), freq j = d (mod 64); exact trig (angles up to 2047) ---
#pragma unroll
  for (int dc = 0; dc < 4; ++dc) {
    const int j = dc * 16 + r;                       // 0..63
    const float invf = powf(10000.f, -(float)j * (1.f / 64.f));
#pragma unroll
    for (int i = 0; i < 8; ++i) {
      const float tpos = (float)(t0 + ihalf + i);
      const float ang  = tpos * invf;
      const float c = cosf(ang), s = sinf(ang);
      const float lo = acc[dc][i], hi = acc[dc + 4][i];
      acc[dc][i]     = (lo * c - hi * s) * QSCALE;   // fold 1/sqrt(Dk)
      acc[dc + 4][i] = (hi * c + lo * s) * QSCALE;
    }
  }

  // --- store Q tile, head-major: q[h][t][d] ---
  float* qb = qout + (size_t)h * T_TOK * D_K;
#pragma unroll
  for (int dc = 0; dc < 8; ++dc)
#pragma unroll
    for (int i = 0; i < 8; ++i)
      qb[(size_t)(t0 + ihalf + i) * D_K + dc * 16 + r] = acc[dc][i];
}

// ---------------------------------------------------------------------------
// Kernel 2: causal flash attention against the KV cache (GQA 2:1).
// One wave per (head, 16-token block); online softmax; P via LDS round-trip.
// ---------------------------------------------------------------------------
__global__ void __launch_bounds__(32)
attn_kernel(const float* __restrict__ q, const float* __restrict__ kcache,
            const float* __restrict__ vcache, float* __restrict__ attn) {
  __shared__ float pbuf[256];
  const int lane  = threadIdx.x;
  const int tb    = blockIdx.x;
  const int h     = blockIdx.y;
  const int t0    = tb * 16;
  const int r     = lane & 15;
  const int koff  = (lane >> 4) * 2;
  const int ihalf = (lane >> 4) * 8;

  const float* Kc = kcache + (size_t)(h >> 1) * S_CACHE * D_K;
  const float* Vc = vcache + (size_t)(h >> 1) * S_CACHE * D_K;
  const float* Qb = q + (size_t)h * T_TOK * D_K;

  // Preload Q A-fragments for this 16-token block (16 x 128 -> 32 chunks)
  v2f af[32];
#pragma unroll
  for (int kc = 0; kc < 32; ++kc)
    af[kc] = *(const v2f*)(Qb + (size_t)(t0 + r) * D_K + kc * 4 + koff);

  v8f o[8] = {};
  float m[8], l[8];
#pragma unroll
  for (int i = 0; i < 8; ++i) { m[i] = -3.0e38f; l[i] = 0.f; }

  for (int sb = 0; sb <= tb; ++sb) {
    const int s0 = sb * 16;

    // scores = Q * K^T  (already includes 1/sqrt(Dk) via Q)
    v8f sc = {};
#pragma unroll 8
    for (int kc = 0; kc < 32; ++kc) {
      v2f b = *(const v2f*)(Kc + (size_t)(s0 + r) * D_K + kc * 4 + koff);
      sc = wmma_f32(af[kc], b, sc);
    }

    // online softmax per row; causal mask only on the diagonal block
    float p[8], alpha[8];
#pragma unroll
    for (int i = 0; i < 8; ++i) {
      float s = sc[i];
      if (sb == tb && (s0 + r) > (t0 + ihalf + i)) s = -3.0e38f;
      const float mx = rowmax16(s);
      const float mn = fmaxf(m[i], mx);
      alpha[i] = __expf(m[i] - mn);
      const float pe = __expf(s - mn);      // masked -> exp(-3e38-mn) == 0
      l[i] = l[i] * alpha[i] + rowsum16(pe);
      m[i] = mn;
      p[i] = pe;
    }

    // D-layout -> A-fragment conversion for P through LDS
#pragma unroll
    for (int i = 0; i < 8; ++i) pbuf[(i + ihalf) * 16 + r] = p[i];
    __syncthreads();
    v2f pf[4];
#pragma unroll
    for (int kc = 0; kc < 4; ++kc) {
      pf[kc].x = pbuf[r * 16 + kc * 4 + koff];
      pf[kc].y = pbuf[r * 16 + kc * 4 + koff + 1];
    }
    __syncthreads();

    // rescale accumulators, then O += P * V
#pragma unroll
    for (int dc = 0; dc < 8; ++dc)
#pragma unroll
      for (int i = 0; i < 8; ++i) o[dc][i] *= alpha[i];

#pragma unroll
    for (int kc = 0; kc < 4; ++kc) {
#pragma unroll
      for (int dc = 0; dc < 8; ++dc) {
        v2f b;
        b.x = Vc[(size_t)(s0 + kc * 4 + koff) * D_K + dc * 16 + r];
        b.y = Vc[(size_t)(s0 + kc * 4 + koff + 1) * D_K + dc * 16 + r];
        o[dc] = wmma_f32(pf[kc], b, o[dc]);
      }
    }
  }

  // finalize and write [t][h*128 + d] (token-major for the output GEMM)
  float linv[8];
#pragma unroll
  for (int i = 0; i < 8; ++i) linv[i] = 1.f / l[i];
#pragma unroll
  for (int dc = 0; dc < 8; ++dc)
#pragma unroll
    for (int i = 0; i < 8; ++i)
      attn[(size_t)(t0 + ihalf + i) * DMODEL + h * D_K + dc * 16 + r] =
          o[dc][i] * linv[i];
}

// ---------------------------------------------------------------------------
// Kernel 3: y = attn @ wo^T
// ---------------------------------------------------------------------------
__global__ void __launch_bounds__(32)
oproj_kernel(const float* __restrict__ a, const float* __restrict__ wo,
             float* __restrict__ y) {
  const int lane  = threadIdx.x;
  const int tb    = blockIdx.x;
  const int nb    = blockIdx.y;            // 128-col output chunk
  const int t0    = tb * 16;
  const int n0    = nb * 128;
  const int r     = lane & 15;
  const int koff  = (lane >> 4) * 2;
  const int ihalf = (lane >> 4) * 8;

  v8f acc[8] = {};
  const float* arow = a + (size_t)(t0 + r) * DMODEL;
  for (int kc = 0; kc < DMODEL / 4; ++kc) {
    const int k = kc * 4 + koff;
    v2f av = *(const v2f*)(arow + k);
#pragma unroll
    for (int dc = 0; dc < 8; ++dc) {
      v2f b = *(const v2f*)(wo + (size_t)(n0 + dc * 16 + r) * DMODEL + k);
      acc[dc] = wmma_f32(av, b, acc[dc]);
    }
  }
#pragma unroll
  for (int dc = 0; dc < 8; ++dc)
#pragma unroll
    for (int i = 0; i < 8; ++i)
      y[(size_t)(t0 + ihalf + i) * DMODEL + n0 + dc * 16 + r] = acc[dc][i];
}

// ---------------------------------------------------------------------------
extern "C" void kernel_launch(void* const* d_in, const int* in_sizes, int n_in,
                              void* d_out, int out_size, void* d_ws,
                              size_t ws_size, hipStream_t stream) {
  (void)in_sizes; (void)n_in; (void)out_size; (void)ws_size;
  const float* x   = (const float*)d_in[0];
  const float* ck  = (const float*)d_in[1];
  const float* cv  = (const float*)d_in[2];
  const float* wq  = (const float*)d_in[3];
  // d_in[4] = wk, d_in[5] = wv, d_in[8] = k_norm_w: provably unused — the
  // causal mask (s <= t, t < 2048) excludes every concatenated-new position
  // (s >= 2048), and exp(-1e30 - max) == 0 in fp32.
  const float* wo  = (const float*)d_in[6];
  const float* qnw = (const float*)d_in[7];
  float* out = (float*)d_out;

  float* q_ws = (float*)d_ws;                               // 16 MB
  float* a_ws = q_ws + (size_t)H_Q * T_TOK * D_K;           // 16 MB

  dim3 blk(32);
  qproj_kernel<<<dim3(T_TOK / 16, H_Q), blk, 0, stream>>>(x, wq, qnw, q_ws);
  attn_kernel<<<dim3(T_TOK / 16, H_Q), blk, 0, stream>>>(q_ws, ck, cv, a_ws);
  oproj_kernel<<<dim3(T_TOK / 16, DMODEL / 128), blk, 0, stream>>>(a_ws, wo, out);
}